// ShortConv_30597347016908
// MI455X (gfx1250) — compile-verified
//
#include <hip/hip_runtime.h>

// Causal depthwise conv1d: y[b,t,d] = bias[d] + sum_{k=0..3} w[d,k] * x[b, t+k-3, d]
// Memory-bound streaming kernel: 128-bit NT loads/stores, register history along T,
// gfx1250 global_prefetch, wave32-coalesced channel-last access.

typedef float v4f __attribute__((ext_vector_type(4)));

constexpr int NB      = 4;     // batch
constexpr int T       = 4096;  // time
constexpr int D       = 2048;  // channels
constexpr int TT      = 64;    // time steps per thread (chunk)
constexpr int THREADS = 256;   // 8 wave32s per block
constexpr int CHB     = THREADS * 4;  // 1024 channels per block
constexpr int DGRP    = D / CHB;      // 2 channel groups

__global__ __launch_bounds__(THREADS)
void ShortConv_stream_kernel(const float* __restrict__ x,
                             const float* __restrict__ w,
                             const float* __restrict__ bias,
                             float* __restrict__ y)
{
    const int d0 = (blockIdx.y * THREADS + threadIdx.x) * 4;  // first of 4 owned channels
    const int t0 = blockIdx.x * TT;                           // chunk start in time
    const int b  = blockIdx.z;

    const float* __restrict__ xp = x + (size_t)b * T * D + d0;
    float*       __restrict__ yp = y + (size_t)b * T * D + d0;

    // Per-channel taps are 4 contiguous floats: w[(d)*4 + k]. Load channel-major,
    // transpose to tap-major vectors across the 4 owned channels.
    const v4f wc0 = *(const v4f*)(w + (size_t)(d0 + 0) * 4);
    const v4f wc1 = *(const v4f*)(w + (size_t)(d0 + 1) * 4);
    const v4f wc2 = *(const v4f*)(w + (size_t)(d0 + 2) * 4);
    const v4f wc3 = *(const v4f*)(w + (size_t)(d0 + 3) * 4);
    const v4f k0 = {wc0.x, wc1.x, wc2.x, wc3.x};
    const v4f k1 = {wc0.y, wc1.y, wc2.y, wc3.y};
    const v4f k2 = {wc0.z, wc1.z, wc2.z, wc3.z};
    const v4f k3 = {wc0.w, wc1.w, wc2.w, wc3.w};
    const v4f bv = *(const v4f*)(bias + d0);

    // History registers x[t-3], x[t-2], x[t-1]. t0 is a multiple of TT (>=64) or 0,
    // so the causal zero-pad only applies to the t0==0 blocks (uniform branch).
    v4f h0 = {0.f, 0.f, 0.f, 0.f};
    v4f h1 = h0;
    v4f h2 = h0;
    if (t0 >= 3) {
        h0 = __builtin_nontemporal_load((const v4f*)(xp + (size_t)(t0 - 3) * D));
        h1 = __builtin_nontemporal_load((const v4f*)(xp + (size_t)(t0 - 2) * D));
        h2 = __builtin_nontemporal_load((const v4f*)(xp + (size_t)(t0 - 1) * D));
    }

    for (int ii = 0; ii < TT; ii += 8) {
        // gfx1250 prefetch path: speculative, OOB at the tail is silently dropped.
        __builtin_prefetch(xp + (size_t)(t0 + ii + 16) * D, 0, 0);
#pragma unroll
        for (int i = 0; i < 8; ++i) {
            const size_t off = (size_t)(t0 + ii + i) * D;
            const v4f xt = __builtin_nontemporal_load((const v4f*)(xp + off));
            v4f r = bv;
            r += k0 * h0;
            r += k1 * h1;
            r += k2 * h2;
            r += k3 * xt;
            __builtin_nontemporal_store(r, (v4f*)(yp + off));
            h0 = h1; h1 = h2; h2 = xt;
        }
    }
}

extern "C" void kernel_launch(void* const* d_in, const int* in_sizes, int n_in,
                              void* d_out, int out_size, void* d_ws, size_t ws_size,
                              hipStream_t stream)
{
    const float* x    = (const float*)d_in[0];  // (4, 4096, 2048) f32
    const float* w    = (const float*)d_in[1];  // (2048, 1, 4)   f32
    const float* bias = (const float*)d_in[2];  // (2048,)        f32
    float*       y    = (float*)d_out;          // (4, 4096, 2048) f32

    dim3 grid(T / TT, DGRP, NB);   // 64 x 2 x 4 = 512 blocks, 4096 wave32s
    ShortConv_stream_kernel<<<grid, THREADS, 0, stream>>>(x, w, bias, y);
}